// Linear_BBB_GP_39376260170046
// MI455X (gfx1250) — compile-verified
//
#include <hip/hip_runtime.h>

typedef float v2f __attribute__((ext_vector_type(2)));
typedef float v8f __attribute__((ext_vector_type(8)));
typedef unsigned int u32x4 __attribute__((ext_vector_type(4)));
typedef int i32x4 __attribute__((ext_vector_type(4)));
typedef int i32x8 __attribute__((ext_vector_type(8)));

constexpr int S_TOT  = 8;
constexpr int SPAIR  = 4;      // sample pairs (p, p+4)
constexpr int BATCH  = 256;
constexpr int KIN    = 2048;
constexpr int NOUT   = 2048;
constexpr int WG_N   = 32;     // output columns per workgroup
constexpr int KT     = 16;     // K tile per iteration
constexpr int XPAD   = 20;     // LDS row stride (floats): 16 data + 4 pad (matches TDM pad)
constexpr unsigned H_W = 16777216u; // S*OUT*IN/2
constexpr unsigned H_B = 8192u;     // S*OUT/2

#if __has_builtin(__builtin_amdgcn_tensor_load_to_lds) && __has_builtin(__builtin_amdgcn_s_wait_tensorcnt)
#define USE_TDM 1
#else
#define USE_TDM 0
#endif

// ---- JAX threefry2x32 (20 rounds), exact ----
__device__ __forceinline__ void tf2x32(unsigned k0, unsigned k1, unsigned x0, unsigned x1,
                                       unsigned& o0, unsigned& o1) {
  const unsigned ks2 = k0 ^ k1 ^ 0x1BD11BDAu;
  x0 += k0; x1 += k1;
  auto r4 = [&](int r0, int r1, int r2, int r3) {
    x0 += x1; x1 = (x1 << r0) | (x1 >> (32 - r0)); x1 ^= x0;
    x0 += x1; x1 = (x1 << r1) | (x1 >> (32 - r1)); x1 ^= x0;
    x0 += x1; x1 = (x1 << r2) | (x1 >> (32 - r2)); x1 ^= x0;
    x0 += x1; x1 = (x1 << r3) | (x1 >> (32 - r3)); x1 ^= x0;
  };
  r4(13, 15, 26, 6);  x0 += k1;  x1 += ks2 + 1u;
  r4(17, 29, 16, 24); x0 += ks2; x1 += k0 + 2u;
  r4(13, 15, 26, 6);  x0 += k0;  x1 += k1 + 3u;
  r4(17, 29, 16, 24); x0 += k1;  x1 += ks2 + 4u;
  r4(13, 15, 26, 6);  x0 += ks2; x1 += k0 + 5u;
  o0 = x0; o1 = x1;
}

// ---- XLA f32 erfinv (Giles) ----
__device__ __forceinline__ float erfinv_f(float x) {
  float w = -log1pf(-x * x);
  float p;
  if (w < 5.0f) {
    w -= 2.5f;
    p = 2.81022636e-08f;
    p = fmaf(p, w, 3.43273939e-07f);
    p = fmaf(p, w, -3.5233877e-06f);
    p = fmaf(p, w, -4.39150654e-06f);
    p = fmaf(p, w, 0.00021858087f);
    p = fmaf(p, w, -0.00125372503f);
    p = fmaf(p, w, -0.00417768164f);
    p = fmaf(p, w, 0.246640727f);
    p = fmaf(p, w, 1.50140941f);
  } else {
    w = __fsqrt_rn(w) - 3.0f;
    p = -0.000200214257f;
    p = fmaf(p, w, 0.000100950558f);
    p = fmaf(p, w, 0.00134934322f);
    p = fmaf(p, w, -0.00367342844f);
    p = fmaf(p, w, 0.00573950773f);
    p = fmaf(p, w, -0.0076224613f);
    p = fmaf(p, w, 0.00943887047f);
    p = fmaf(p, w, 1.00167406f);
    p = fmaf(p, w, 2.83297682f);
  }
  return p * x;
}

__device__ __forceinline__ float bits_to_normal(unsigned b) {
  unsigned fb = (b >> 9) | 0x3f800000u;
  float f = __uint_as_float(fb) - 1.0f;          // [0,1)
  float u = fmaf(f, 2.0f, -0.99999994f);         // JAX uniform(-1+ulp, 1)
  return 1.41421356f * erfinv_f(u);
}

__device__ __forceinline__ float softplus_f(float x) {
  return fmaxf(x, 0.0f) + log1pf(__expf(-fabsf(x)));
}

#if USE_TDM
// Issue a 2D TDM load: tile 16 f32 x 256 rows, row stride KIN elems, LDS rows padded 16+4 dwords.
__device__ __forceinline__ void tdm_load_x_tile(unsigned lds_off, unsigned long long ga) {
  u32x4 g0;
  g0.x = 1u;                                                   // count=1 (valid), gather off
  g0.y = lds_off;                                              // LDS byte address
  g0.z = (unsigned)ga;                                         // global addr [31:0]
  g0.w = (unsigned)((ga >> 32) & 0x01FFFFFFull) | (2u << 30);  // addr[56:32] | type=2
  i32x8 g1;
  g1[0] = (int)((2u << 16) | (1u << 20) | (3u << 22) | (3u << 25)); // 4B, pad_en, every 16dw pad 4dw
  g1[1] = (int)(16u << 16);          // tensor_dim0 = 16
  g1[2] = (int)(256u << 16);         // tensor_dim0 hi=0, tensor_dim1 = 256
  g1[3] = (int)(16u << 16);          // tensor_dim1 hi=0, tile_dim0 = 16
  g1[4] = (int)256u;                 // tile_dim1 = 256, tile_dim2 = 0
  g1[5] = (int)2048u;                // tensor_dim0_stride = 2048 (elems)
  g1[6] = 0; g1[7] = 0;
  i32x4 gz = {0, 0, 0, 0};
#if defined(__clang_major__) && (__clang_major__ >= 23)
  i32x8 gz8 = {0, 0, 0, 0, 0, 0, 0, 0};
  __builtin_amdgcn_tensor_load_to_lds(g0, g1, gz, gz, gz8, 0);
#else
  __builtin_amdgcn_tensor_load_to_lds(g0, g1, gz, gz, 0);
#endif
}
#endif

__global__ __launch_bounds__(256) void bbb_wmma_f32_kernel(
    const float* __restrict__ x, const float* __restrict__ wmu,
    const float* __restrict__ wrho, const float* __restrict__ bmu,
    const float* __restrict__ brho, float* __restrict__ out) {
  __shared__ float Xs[2][BATCH][XPAD];   // staged x tiles, both samples of pair
  __shared__ float Ws[2][WG_N][XPAD];    // generated W tiles

  const int tid   = threadIdx.x;
  const int wave  = tid >> 5;
  const int lane  = tid & 31;
  const int lhalf = lane >> 4;
  const int l16   = lane & 15;

  const int p  = blockIdx.y;             // sample pair index: samples p and p+4
  const int o0 = blockIdx.x * WG_N;      // output column base

  // Derive kw, kb from jax.random.key(42) split (key data = {0,42})
  unsigned kw0, kw1, kb0, kb1, t0, t1;
  tf2x32(0u, 42u, 0u, 2u, t0, t1); kw0 = t0; kb0 = t1;
  tf2x32(0u, 42u, 1u, 3u, t0, t1); kw1 = t0; kb1 = t1;

  v8f acc[2][2][2];                      // [sample][m-tile][n-tile]
  const v8f vz = {0.f, 0.f, 0.f, 0.f, 0.f, 0.f, 0.f, 0.f};
#pragma unroll
  for (int sp = 0; sp < 2; ++sp)
#pragma unroll
    for (int mi = 0; mi < 2; ++mi)
#pragma unroll
      for (int ni = 0; ni < 2; ++ni) acc[sp][mi][ni] = vz;

  const int m_base = wave * 32;          // 2 m-tiles of 16 rows per wave
  const float* xs[2] = {x + (size_t)p * BATCH * KIN,
                        x + (size_t)(p + SPAIR) * BATCH * KIN};
#if USE_TDM
  const unsigned lds_xoff[2] = {(unsigned)(unsigned long long)(uintptr_t)(void*)&Xs[0][0][0],
                                (unsigned)(unsigned long long)(uintptr_t)(void*)&Xs[1][0][0]};
#endif

  for (int k0 = 0; k0 < KIN; k0 += KT) {
    __syncthreads();
#if USE_TDM
    // ---- async DMA of X tiles into LDS via Tensor Data Mover (wave 0 only) ----
    if (wave == 0) {
#pragma unroll
      for (int sp = 0; sp < 2; ++sp) {
        unsigned long long ga =
            (unsigned long long)(uintptr_t)(const void*)xs[sp] + (unsigned long long)k0 * 4ull;
        tdm_load_x_tile(lds_xoff[sp], ga);
      }
    }
#else
    // ---- fallback: manual staging (coalesced b128) ----
#pragma unroll
    for (int sp = 0; sp < 2; ++sp) {
      const float4* gx = (const float4*)(xs[sp] + (size_t)tid * KIN + k0);
#pragma unroll
      for (int j = 0; j < 4; ++j) *(float4*)&Xs[sp][tid][4 * j] = gx[j];
    }
#endif
    // ---- generate W tile: 32 outs x 16 k; one threefry per element covers both samples ----
#pragma unroll
    for (int q = 0; q < 2; ++q) {
      const int pos = tid + q * 256;     // 0..511
      const int ol  = pos >> 4;          // 0..31
      const int kk  = pos & 15;
      const int o   = o0 + ol;
      const int i   = k0 + kk;
      const size_t gidx = (size_t)o * KIN + i;
      const float mu = wmu[gidx];
      const float sg = softplus_f(wrho[gidx]);
      const unsigned c0 = (unsigned)(p * NOUT * KIN) + (unsigned)(o * KIN) + (unsigned)i;
      unsigned r0, r1;
      tf2x32(kw0, kw1, c0, c0 + H_W, r0, r1);
      Ws[0][ol][kk] = fmaf(sg, bits_to_normal(r0), mu);
      Ws[1][ol][kk] = fmaf(sg, bits_to_normal(r1), mu);
    }
#if USE_TDM
    if (wave == 0) __builtin_amdgcn_s_wait_tensorcnt(0);
#endif
    __syncthreads();

    // ---- 4 WMMA sub-steps of K=4 ----
#pragma unroll
    for (int sub = 0; sub < 4; ++sub) {
      const int c = 4 * sub + 2 * lhalf; // K pair per half-wave per ISA 16x4/4x16 layout
      v2f af[2][2], bf[2][2];
#pragma unroll
      for (int sp = 0; sp < 2; ++sp) {
#pragma unroll
        for (int mi = 0; mi < 2; ++mi)
          af[sp][mi] = *(const v2f*)&Xs[sp][m_base + 16 * mi + l16][c];
#pragma unroll
        for (int ni = 0; ni < 2; ++ni)
          bf[sp][ni] = *(const v2f*)&Ws[sp][16 * ni + l16][c];
      }
#pragma unroll
      for (int sp = 0; sp < 2; ++sp)
#pragma unroll
        for (int mi = 0; mi < 2; ++mi)
#pragma unroll
          for (int ni = 0; ni < 2; ++ni)
            acc[sp][mi][ni] = __builtin_amdgcn_wmma_f32_16x16x4_f32(
                false, af[sp][mi], false, bf[sp][ni], (short)0,
                acc[sp][mi][ni], false, false);
    }
  }

  // ---- bias: one threefry per column gives eps_b for both paired samples ----
  float biasv[2][2];
#pragma unroll
  for (int ni = 0; ni < 2; ++ni) {
    const int o = o0 + 16 * ni + l16;
    const float mu = bmu[o];
    const float sg = softplus_f(brho[o]);
    const unsigned c0 = (unsigned)(p * NOUT + o);
    unsigned r0, r1;
    tf2x32(kb0, kb1, c0, c0 + H_B, r0, r1);
    biasv[0][ni] = fmaf(sg, bits_to_normal(r0), mu);
    biasv[1][ni] = fmaf(sg, bits_to_normal(r1), mu);
  }

  // ---- store: D layout VGPR r -> row m_base+16*mi+8*lhalf+r, col o0+16*ni+l16 ----
#pragma unroll
  for (int sp = 0; sp < 2; ++sp) {
    float* op = out + (size_t)(p + SPAIR * sp) * BATCH * NOUT;
#pragma unroll
    for (int mi = 0; mi < 2; ++mi)
#pragma unroll
      for (int ni = 0; ni < 2; ++ni) {
        const int ocol = o0 + 16 * ni + l16;
#pragma unroll
        for (int r = 0; r < 8; ++r) {
          const int brow = m_base + 16 * mi + 8 * lhalf + r;
          op[(size_t)brow * NOUT + ocol] = acc[sp][mi][ni][r] + biasv[sp][ni];
        }
      }
  }
}

extern "C" void kernel_launch(void* const* d_in, const int* in_sizes, int n_in,
                              void* d_out, int out_size, void* d_ws, size_t ws_size,
                              hipStream_t stream) {
  (void)in_sizes; (void)n_in; (void)out_size; (void)d_ws; (void)ws_size;
  const float* x    = (const float*)d_in[0];
  const float* wmu  = (const float*)d_in[1];
  const float* wrho = (const float*)d_in[2];
  const float* bmu  = (const float*)d_in[3];
  const float* brho = (const float*)d_in[4];
  float* out = (float*)d_out;
  dim3 grid(NOUT / WG_N, SPAIR);   // 64 x 4 workgroups
  dim3 block(256);                 // 8 waves (wave32)
  bbb_wmma_f32_kernel<<<grid, block, 0, stream>>>(x, wmu, wrho, bmu, brho, out);
}